// PointsRasterizerScanNet_16131897164118
// MI455X (gfx1250) — compile-verified
//
#include <hip/hip_runtime.h>
#include <hip/hip_bf16.h>

typedef __attribute__((ext_vector_type(2))) float v2f;
typedef __attribute__((ext_vector_type(8))) float v8f;

#define IMG    64
#define NPIX   (IMG * IMG)          // 4096
#define KTOP   8
#define NPTS   8192
#define NB     2
#define RAD2   0.0025f              // 0.05^2
#define ZINF   1.0e10f
#define CHUNK  512                  // points staged in LDS per outer step
#define SWAPX16 0x401F              // ds_swizzle: xor=0x10, or=0, and=0x1f

// ---------------------------------------------------------------------------
// Kernel 1: project points -> float4(ax, ay, s, z) table in workspace.
//   ax = -2*sx, ay = -2*sy, s = sx^2+sy^2 (or 1e10 sentinel if z<=0), z = depth
//   dist2(pixel,point) = ax*px + ay*py + s*1 + 1*(px^2+py^2)  (a K=4 dot)
// ---------------------------------------------------------------------------
__global__ __launch_bounds__(256) void pr_project(
    const float* __restrict__ points, const float* __restrict__ R,
    const float* __restrict__ T, const float* __restrict__ focal,
    float4* __restrict__ sp)
{
    int t = blockIdx.x * 256 + threadIdx.x;
    if (t >= NB * NPTS) return;
    int b = t / NPTS;
    const float* p  = points + (size_t)t * 3;
    const float* Rm = R + b * 9;   // row-major [3][3]
    const float* Tv = T + b * 3;
    float p0 = p[0], p1 = p[1], p2 = p[2];
    // row-vector convention: view_j = sum_i p_i * R[i][j] + T_j
    float v0 = p0 * Rm[0] + p1 * Rm[3] + p2 * Rm[6] + Tv[0];
    float v1 = p0 * Rm[1] + p1 * Rm[4] + p2 * Rm[7] + Tv[1];
    float z  = p0 * Rm[2] + p1 * Rm[5] + p2 * Rm[8] + Tv[2];
    float4 o;
    if (z > 0.0f) {
        float sx = -(focal[0] * v0 / z);
        float sy = -(focal[1] * v1 / z);
        o.x = -2.0f * sx;
        o.y = -2.0f * sy;
        o.z = sx * sx + sy * sy;
    } else {
        o.x = 0.0f; o.y = 0.0f; o.z = ZINF;   // can never pass radius test
    }
    o.w = z;
    sp[t] = o;
}

// Branch-free sorted insert of u64 key (z_bits<<32 | idx); caller guarantees
// c < kq[7].  Lowers to v_min_u64 / v_max_u64 chains -- no list copies.
__device__ __forceinline__ void ins8(unsigned long long c, unsigned long long kq[KTOP])
{
    kq[KTOP - 1] = c;
#pragma unroll
    for (int j = KTOP - 1; j > 0; --j) {
        unsigned long long a = kq[j - 1], b = kq[j];
        bool lt = a < b;
        kq[j - 1] = lt ? a : b;
        kq[j]     = lt ? b : a;
    }
}

// ---------------------------------------------------------------------------
// Kernel 2: rasterize. 256 threads = 8 waves; each wave owns 16 pixels.
// One V_WMMA_F32_16X16X4_F32 per 16-point chunk computes the 16x16 dist^2
// tile.  C layout: lane%16 = pixel (N), VGPR r = point (M; +8 for lanes>=16).
// LDS staging uses global_load_async_to_lds_b128 (ASYNCcnt-tracked).
// ---------------------------------------------------------------------------
__global__ __launch_bounds__(256) void pr_raster(
    const float4* __restrict__ sp,
    int* __restrict__ out_idx, float* __restrict__ out_z, float* __restrict__ out_d)
{
    __shared__ float4 lds[CHUNK];   // 8 KB of the 320 KB/WGP LDS

    int cloud  = blockIdx.x >> 5;            // 32 blocks per cloud
    int pixblk = (blockIdx.x & 31) * 128;    // 8 waves * 16 pixels
    int tid  = threadIdx.x;
    int lane = tid & 31;
    int wave = tid >> 5;
    int n    = lane & 15;                    // pixel slot within wave tile
    int half = lane >> 4;                    // which 8-point half of M
    int pixel = pixblk + wave * 16 + n;

    float px = 1.0f - (2.0f * (float)(pixel & 63) + 1.0f) / 64.0f;
    float py = 1.0f - (2.0f * (float)(pixel >> 6) + 1.0f) / 64.0f;

    // B operand 4x16: lanes 0-15 hold K=0/1 rows, lanes 16-31 hold K=2/3.
    v2f bop;
    bop[0] = half ? 1.0f : px;
    bop[1] = half ? (px * px + py * py) : py;

    unsigned long long kq[KTOP];
#pragma unroll
    for (int k = 0; k < KTOP; ++k) kq[k] = ~0ULL;

    const float4* base = sp + (size_t)cloud * NPTS;

    for (int outer = 0; outer < NPTS; outer += CHUNK) {
        __syncthreads();                     // previous chunk fully consumed
#pragma unroll
        for (int l = 0; l < CHUNK / 256; ++l) {
            unsigned lds_off = (unsigned)(uintptr_t)(&lds[tid + l * 256]);
            unsigned long long ga =
                (unsigned long long)(uintptr_t)(&base[outer + tid + l * 256]);
            asm volatile("global_load_async_to_lds_b128 %0, %1, off"
                         :: "v"(lds_off), "v"(ga) : "memory");
        }
        asm volatile("s_wait_asynccnt 0x0" ::: "memory");
        __syncthreads();                     // chunk visible to all waves

        for (int c0 = 0; c0 < CHUNK; c0 += 16) {
            // A operand 16x4: lane%16 = point row M; lanes 0-15: K=0/1,
            // lanes 16-31: K=2/3.
            float4 a4 = lds[c0 + n];
            v2f aop;
            aop[0] = half ? a4.z : a4.x;
            aop[1] = half ? 1.0f : a4.y;
            v8f acc = {};
            v8f d = __builtin_amdgcn_wmma_f32_16x16x4_f32(
                false, aop, false, bop, (short)0, acc, false, false);

            // Single wave-level gate: min-reduce the 8 distances so the
            // common (no-candidate) path carries no top-8 list merges.
            float dmin = fminf(fminf(fminf(d[0], d[1]), fminf(d[2], d[3])),
                               fminf(fminf(d[4], d[5]), fminf(d[6], d[7])));
            if (dmin <= RAD2) {                       // rare (~2% of chunks)
#pragma unroll
                for (int r = 0; r < 8; ++r) {
                    if (d[r] <= RAD2) {
                        int li = c0 + half * 8 + r;
                        float z = lds[li].w;          // broadcast ds read
                        unsigned long long c =
                            ((unsigned long long)__float_as_uint(z) << 32) |
                            (unsigned)(outer + li);
                        if (c < kq[KTOP - 1]) ins8(c, kq);
                    }
                }
            }
        }
    }

    // Merge lane p with lane p^16 (disjoint point halves, same pixel).
    // Snapshot partner list first (uniform EXEC), then insert.
    unsigned long long pk[KTOP];
#pragma unroll
    for (int k = 0; k < KTOP; ++k) {
        unsigned lo = __builtin_amdgcn_ds_swizzle((int)(unsigned)kq[k], SWAPX16);
        unsigned hi = __builtin_amdgcn_ds_swizzle((int)(unsigned)(kq[k] >> 32), SWAPX16);
        pk[k] = ((unsigned long long)hi << 32) | lo;
    }
#pragma unroll
    for (int k = 0; k < KTOP; ++k)
        if (pk[k] < kq[KTOP - 1]) ins8(pk[k], kq);

    if (half == 0) {
        size_t ob = ((size_t)cloud * NPIX + (size_t)pixel) * KTOP;
#pragma unroll
        for (int k = 0; k < KTOP; ++k) {
            unsigned hi = (unsigned)(kq[k] >> 32);
            unsigned lo = (unsigned)kq[k];
            float zf = __uint_as_float(hi);
            int oi; float oz, od;
            if (zf < 0.5f * ZINF) {          // valid (NaN/sentinel fails)
                float4 q = base[lo];         // L2 hit; <=8 per pixel
                float sx = -0.5f * q.x, sy = -0.5f * q.y;
                float dx = px - sx, dy = py - sy;
                oi = (int)lo; oz = zf; od = dx * dx + dy * dy;  // exact dist2
            } else {
                oi = -1; oz = -1.0f; od = -1.0f;
            }
            out_idx[ob + k] = oi;
            out_z[ob + k]   = oz;
            out_d[ob + k]   = od;
        }
    }
}

extern "C" void kernel_launch(void* const* d_in, const int* in_sizes, int n_in,
                              void* d_out, int out_size, void* d_ws, size_t ws_size,
                              hipStream_t stream) {
    const float* points = (const float*)d_in[0];   // [2,8192,3]
    const float* R      = (const float*)d_in[1];   // [2,3,3]
    const float* T      = (const float*)d_in[2];   // [2,3]
    const float* focal  = (const float*)d_in[3];   // [2]

    float4* sp = (float4*)d_ws;                    // 256 KB point table

    pr_project<<<(NB * NPTS + 255) / 256, 256, 0, stream>>>(points, R, T, focal, sp);

    const size_t seg = (size_t)NB * NPIX * KTOP;   // 65536 elements per output
    int*   out_idx = (int*)d_out;                  // idx written as int32 bits
    float* out_z   = (float*)d_out + seg;
    float* out_d   = out_z + seg;

    pr_raster<<<NB * (NPIX / 128), 256, 0, stream>>>(sp, out_idx, out_z, out_d);
}